// GATModel_25048249270535
// MI455X (gfx1250) — compile-verified
//
#include <hip/hip_runtime.h>
#include <math.h>

#define N_NODES 100000
#define N_EDGES 1600000
#define D 64
#define ED 16
#define N_LAYERS 3
#define NEG_SLOPE 0.2f
#define ETOT (N_EDGES + N_NODES)

typedef float v2f __attribute__((ext_vector_type(2)));
typedef float v8f __attribute__((ext_vector_type(8)));

// ---------------- WMMA GEMM: h[N,64] = x[N,64] @ W[64,64] ----------------
// One wave computes one 16x16 output tile via 16 chained V_WMMA_F32_16X16X4_F32.
// Block = 256 threads = 8 waves = 2 row-tiles x 4 col-tiles = 32 rows.
__global__ __launch_bounds__(256) void k_gemm(const float* __restrict__ x,
                                              const float* __restrict__ W,
                                              float* __restrict__ h) {
  const int wave = threadIdx.x >> 5;
  const int lane = threadIdx.x & 31;
  const int half = lane >> 4;       // 0: lanes 0-15, 1: lanes 16-31
  const int l16  = lane & 15;
  const int row0 = blockIdx.x * 32 + (wave >> 2) * 16;
  const int col0 = (wave & 3) * 16;

  const int mA = row0 + l16;        // A: lane holds row M=l16, K pair per half
  const int n  = col0 + l16;        // B/C/D: lane holds column N=l16

  v8f c = {};
#pragma unroll
  for (int kk = 0; kk < 64; kk += 4) {
    const int kA = kk + half * 2;
    v2f a = *(const v2f*)(x + mA * 64 + kA);     // A[m][kA], A[m][kA+1]
    v2f b;
    b.x = W[kA * 64 + n];                        // B[kA][n]
    b.y = W[(kA + 1) * 64 + n];                  // B[kA+1][n]
    c = __builtin_amdgcn_wmma_f32_16x16x4_f32(false, a, false, b,
                                              (short)0, c, false, false);
  }
  // C/D layout: VGPR r -> row M = r + half*8, col N = l16
#pragma unroll
  for (int r = 0; r < 8; ++r)
    h[(row0 + r + half * 8) * 64 + n] = c[r];
}

// ---------------- per-node attention scores ----------------
__global__ void k_scores(const float* __restrict__ h, const float* __restrict__ a_src,
                         const float* __restrict__ a_dst, float* __restrict__ ssrc,
                         float* __restrict__ sdst) {
  int i = blockIdx.x * blockDim.x + threadIdx.x;
  if (i >= N_NODES) return;
  const float4* hp = (const float4*)(h + i * 64);
  float s = 0.f, t = 0.f;
#pragma unroll
  for (int j = 0; j < 16; ++j) {
    float4 v = hp[j];
    s += v.x * a_src[4*j] + v.y * a_src[4*j+1] + v.z * a_src[4*j+2] + v.w * a_src[4*j+3];
    t += v.x * a_dst[4*j] + v.y * a_dst[4*j+1] + v.z * a_dst[4*j+2] + v.w * a_dst[4*j+3];
  }
  ssrc[i] = s;
  sdst[i] = t;
}

// ---------------- per-layer init: agg=0, segmax=-inf, segsum=0 ----------------
__global__ void k_init(float* __restrict__ agg, float* __restrict__ m, float* __restrict__ z) {
  int i = blockIdx.x * blockDim.x + threadIdx.x;
  if (i < N_NODES * D) agg[i] = 0.f;
  if (i < N_NODES) { m[i] = -INFINITY; z[i] = 0.f; }
}

__global__ void k_zero_small(float* p, int n) {
  int i = threadIdx.x;
  if (i < n) p[i] = 0.f;
}

// ---------------- edge_attr column sums (once; mean only enters via a scalar) ----
__global__ __launch_bounds__(256) void k_easum(const float* __restrict__ ea,
                                               float* __restrict__ ea_sum) {
  __shared__ float red[256 * ED];
  float acc[ED];
#pragma unroll
  for (int c = 0; c < ED; ++c) acc[c] = 0.f;
  for (long long e = (long long)blockIdx.x * blockDim.x + threadIdx.x; e < N_EDGES;
       e += (long long)gridDim.x * blockDim.x) {
    const float4* p = (const float4*)(ea + e * ED);
#pragma unroll
    for (int j = 0; j < 4; ++j) {
      float4 v = p[j];
      acc[4*j] += v.x; acc[4*j+1] += v.y; acc[4*j+2] += v.z; acc[4*j+3] += v.w;
    }
  }
#pragma unroll
  for (int c = 0; c < ED; ++c) red[threadIdx.x * ED + c] = acc[c];
  __syncthreads();
  for (int off = 128; off > 0; off >>= 1) {
    if (threadIdx.x < off) {
#pragma unroll
      for (int c = 0; c < ED; ++c)
        red[threadIdx.x * ED + c] += red[(threadIdx.x + off) * ED + c];
    }
    __syncthreads();
  }
  if (threadIdx.x < ED) atomicAdd(&ea_sum[threadIdx.x], red[threadIdx.x]);
}

// ---------------- per-layer tiny prep: ve = W_e @ a_e ; sloop = mean_ea . ve ----
__global__ void k_prep(const float* __restrict__ We, const float* __restrict__ ae,
                       const float* __restrict__ ea_sum, float* __restrict__ ve,
                       float* __restrict__ sloop) {
  __shared__ float sve[ED];
  int t = threadIdx.x;
  if (t < ED) {
    float acc = 0.f;
    for (int d = 0; d < D; ++d) acc += We[t * D + d] * ae[d];
    sve[t] = acc;
    ve[t] = acc;
  }
  __syncthreads();
  if (t == 0) {
    float s = 0.f;
    for (int j = 0; j < ED; ++j) s += ea_sum[j] * sve[j];
    *sloop = s * (1.0f / (float)N_EDGES);
  }
}

// ---------------- edge logits + leaky-relu + segment-max (atomic) ----------------
__global__ void k_alpha(const int* __restrict__ ei, const float* __restrict__ ea,
                        const float* __restrict__ ssrc, const float* __restrict__ sdst,
                        const float* __restrict__ ve, const float* __restrict__ sloop,
                        float* __restrict__ alpha, float* __restrict__ m) {
  int e = blockIdx.x * blockDim.x + threadIdx.x;
  if (e >= ETOT) return;
  int s, d;
  float se;
  if (e < N_EDGES) {
    s = ei[e];
    d = ei[N_EDGES + e];
    const float4* p = (const float4*)(ea + (long long)e * ED);
    float acc = 0.f;
#pragma unroll
    for (int j = 0; j < 4; ++j) {
      float4 v = p[j];
      acc += v.x * ve[4*j] + v.y * ve[4*j+1] + v.z * ve[4*j+2] + v.w * ve[4*j+3];
    }
    se = acc;
  } else {
    s = d = e - N_EDGES;
    se = *sloop;
  }
  float a = ssrc[s] + sdst[d] + se;
  a = (a >= 0.f) ? a : NEG_SLOPE * a;
  alpha[e] = a;
  // float atomic max via signed-max (a>=0) / unsigned-min (a<0); m init'd to -inf
  if (a >= 0.f) atomicMax((int*)(m + d), __float_as_int(a));
  else          atomicMin((unsigned int*)(m + d), __float_as_uint(a));
}

// ---------------- exp(alpha - max) and segment-sum ----------------
__global__ void k_expsum(const int* __restrict__ ei, const float* __restrict__ m,
                         float* __restrict__ alpha, float* __restrict__ z) {
  int e = blockIdx.x * blockDim.x + threadIdx.x;
  if (e >= ETOT) return;
  int d = (e < N_EDGES) ? ei[N_EDGES + e] : (e - N_EDGES);
  float ev = expf(alpha[e] - m[d]);
  alpha[e] = ev;
  atomicAdd(&z[d], ev);
}

// ---------------- weighted scatter-aggregate: one wave per edge ----------------
__global__ __launch_bounds__(256) void k_agg(const int* __restrict__ ei,
                                             const float* __restrict__ alpha,
                                             const float* __restrict__ z,
                                             const float* __restrict__ h,
                                             float* __restrict__ agg) {
  int e = blockIdx.x * 8 + (threadIdx.x >> 5);
  int lane = threadIdx.x & 31;
  if (e >= ETOT) return;
  int s, d;
  if (e < N_EDGES) { s = ei[e]; d = ei[N_EDGES + e]; }
  else             { s = d = e - N_EDGES; }
  float w = alpha[e] / z[d];
  const float* hs = h + (long long)s * 64;
  float* od = agg + (long long)d * 64;
  atomicAdd(od + lane,      w * hs[lane]);
  atomicAdd(od + lane + 32, w * hs[lane + 32]);
}

// ---------------- bias + exact GELU ----------------
__global__ void k_bias_gelu(const float* __restrict__ agg, const float* __restrict__ b,
                            float* __restrict__ out) {
  int i = blockIdx.x * blockDim.x + threadIdx.x;
  if (i >= N_NODES * D) return;
  float v = agg[i] + b[i & 63];
  out[i] = 0.5f * v * (1.0f + erff(v * 0.70710678118654752440f));
}

extern "C" void kernel_launch(void* const* d_in, const int* in_sizes, int n_in,
                              void* d_out, int out_size, void* d_ws, size_t ws_size,
                              hipStream_t stream) {
  (void)in_sizes; (void)n_in; (void)out_size; (void)ws_size;
  const float* x0   = (const float*)d_in[0];
  const int*   ei   = (const int*)d_in[1];    // [2, E]: row0=src, row1=dst
  const float* ea   = (const float*)d_in[2];  // [E, 16]
  const float* Ws   = (const float*)d_in[3];  // [3, 64, 64]
  const float* asrc = (const float*)d_in[4];  // [3, 64]
  const float* adst = (const float*)d_in[5];  // [3, 64]
  const float* We   = (const float*)d_in[6];  // [3, 16, 64]
  const float* ae   = (const float*)d_in[7];  // [3, 64]
  const float* bias = (const float*)d_in[8];  // [3, 64]

  float* ws = (float*)d_ws;
  float* h     = ws; ws += N_NODES * D;   // 25.6 MB
  float* agg   = ws; ws += N_NODES * D;   // 25.6 MB
  float* xpp   = ws; ws += N_NODES * D;   // 25.6 MB
  float* ssrc  = ws; ws += N_NODES;
  float* sdst  = ws; ws += N_NODES;
  float* alph  = ws; ws += ETOT;
  float* mbuf  = ws; ws += N_NODES;
  float* zbuf  = ws; ws += N_NODES;
  float* small = ws;                      // [0..15]=ea_sum, [16..31]=ve, [32]=sloop
  float* ea_sum = small;
  float* ve     = small + 16;
  float* sloop  = small + 32;

  // edge-attr column sums, computed once (layer-independent)
  k_zero_small<<<1, 64, 0, stream>>>(small, 64);
  k_easum<<<1024, 256, 0, stream>>>(ea, ea_sum);

  for (int l = 0; l < N_LAYERS; ++l) {
    const float* x_in = (l == 0) ? x0 : xpp;
    float* x_out = (l == N_LAYERS - 1) ? (float*)d_out : xpp;

    k_init<<<(N_NODES * D + 255) / 256, 256, 0, stream>>>(agg, mbuf, zbuf);
    k_prep<<<1, 64, 0, stream>>>(We + l * ED * D, ae + l * D, ea_sum, ve, sloop);
    k_gemm<<<N_NODES / 32, 256, 0, stream>>>(x_in, Ws + l * D * D, h);
    k_scores<<<(N_NODES + 255) / 256, 256, 0, stream>>>(h, asrc + l * D, adst + l * D,
                                                        ssrc, sdst);
    k_alpha<<<(ETOT + 255) / 256, 256, 0, stream>>>(ei, ea, ssrc, sdst, ve, sloop,
                                                    alph, mbuf);
    k_expsum<<<(ETOT + 255) / 256, 256, 0, stream>>>(ei, mbuf, alph, zbuf);
    k_agg<<<(ETOT + 7) / 8, 256, 0, stream>>>(ei, alph, zbuf, h, agg);
    k_bias_gelu<<<(N_NODES * D + 255) / 256, 256, 0, stream>>>(agg, bias + l * D, x_out);
  }
}